// PerceiverPreprocessor_65377992180271
// MI455X (gfx1250) — compile-verified
//
#include <hip/hip_runtime.h>

typedef float v2f __attribute__((ext_vector_type(2)));
typedef float v8f __attribute__((ext_vector_type(8)));

#define DMODEL 128
#define LDA 132   // padded LDS stride: 132 mod 64 == 4 -> conflict-free WMMA A-frag reads

// ---------------------------------------------------------------- degree prep
__global__ void __launch_bounds__(256) gcn_deg_init(float* __restrict__ deg, int n) {
  int i = blockIdx.x * 256 + threadIdx.x;
  if (i < n) deg[i] = 1.0f;   // self-loop
}

__global__ void __launch_bounds__(256) gcn_deg_accum(const long long* __restrict__ dst,
                                                     float* __restrict__ deg, int e) {
  int i = blockIdx.x * 256 + threadIdx.x;
  if (i < e) unsafeAtomicAdd(&deg[(int)dst[i]], 1.0f);
}

__global__ void __launch_bounds__(256) gcn_dinv(float* __restrict__ deg, int n) {
  int i = blockIdx.x * 256 + threadIdx.x;
  if (i < n) deg[i] = rsqrtf(deg[i]);
}

// ------------------------------------------------- fused GEMM (fp32 WMMA) ----
// hw  = A @ W                      [n x 128]
// agg = hw * dinv^2 + bias         [n x 128]  (self-loop init for scatter-add)
// Block = 256 threads = 8 waves; wave w owns output cols [16w, 16w+16).
// Each wave preloads its 128x16 W slab as 32 f32-WMMA B fragments in VGPRs,
// then grid-strides over 16-row tiles of A staged in LDS.
__global__ void __launch_bounds__(256) gcn_gemm_fused(
    const float* __restrict__ A, const float* __restrict__ W,
    const float* __restrict__ bias, const float* __restrict__ dinv,
    float* __restrict__ hw, float* __restrict__ agg, int n, int ntiles) {
  __shared__ float As[16 * LDA];

  const int tid  = threadIdx.x;
  const int lane = tid & 31;
  const int wave = tid >> 5;
  const int nn   = lane & 15;          // col within tile / A-frag row
  const int kh   = (lane >> 4) * 2;    // K sub-offset: lanes 0-15 -> 0, 16-31 -> 2
  const int c0   = wave * 16;

  // Preload B fragments for the whole K=128 (V_WMMA_F32_16X16X4_F32 layout):
  // b.x = W[k+kh][c0+nn], b.y = W[k+kh+1][c0+nn]
  v2f bfrag[32];
#pragma unroll
  for (int kk = 0; kk < 32; ++kk) {
    int k = kk * 4 + kh;
    bfrag[kk].x = W[(k    ) * DMODEL + c0 + nn];
    bfrag[kk].y = W[(k + 1) * DMODEL + c0 + nn];
  }
  const float bcol = bias[c0 + nn];

  for (int tile = blockIdx.x; tile < ntiles; tile += gridDim.x) {
    const int m0 = tile * 16;

    // Stage 16x128 A tile (zero-fill out-of-range rows).
#pragma unroll
    for (int it = 0; it < 2; ++it) {
      int idx = tid + it * 256;          // 512 float4 slots = 16x128 floats
      int r = idx >> 5;
      int c = (idx & 31) * 4;
      float4 v = make_float4(0.f, 0.f, 0.f, 0.f);
      int row = m0 + r;
      if (row < n) v = *(const float4*)(A + (size_t)row * DMODEL + c);
      *(float4*)(&As[r * LDA + c]) = v;
    }
    __syncthreads();

    v8f acc = {};
#pragma unroll
    for (int kk = 0; kk < 32; ++kk) {
      int k = kk * 4 + kh;
      v2f a = *(const v2f*)(&As[nn * LDA + k]);   // a.x=A[nn][k+kh], a.y=A[nn][k+kh+1]
      acc = __builtin_amdgcn_wmma_f32_16x16x4_f32(
          /*neg_a=*/false, a, /*neg_b=*/false, bfrag[kk],
          /*c_mod=*/(short)0, acc, /*reuse_a=*/false, /*reuse_b=*/false);
    }

    // D layout: VGPR r -> M = r + (lane>=16 ? 8 : 0), N = lane&15.
    const int mbase = m0 + ((lane >> 4) * 8);
#pragma unroll
    for (int r = 0; r < 8; ++r) {
      int m = mbase + r;
      if (m < n) {
        float t  = acc[r];
        float di = dinv[m];
        size_t o = (size_t)m * DMODEL + c0 + nn;
        hw[o]  = t;
        agg[o] = t * di * di + bcol;
      }
    }
    __syncthreads();
  }
}

// --------------------------------------------------------- edge scatter-add --
// One edge per wave32: lane handles 4 columns (32 lanes * 4 = 128).
__global__ void __launch_bounds__(256) gcn_scatter(
    const long long* __restrict__ src, const long long* __restrict__ dst,
    const float* __restrict__ dinv, const float* __restrict__ hw,
    float* __restrict__ agg, int e) {
  unsigned gid  = blockIdx.x * 256u + threadIdx.x;
  unsigned edge = gid >> 5;
  if (edge >= (unsigned)e) return;
  int c = (gid & 31) * 4;
  int s = (int)src[edge];
  int d = (int)dst[edge];
  float norm = dinv[s] * dinv[d];
  float4 v = *(const float4*)(hw + (size_t)s * DMODEL + c);
  float* o = agg + (size_t)d * DMODEL + c;
  unsafeAtomicAdd(o + 0, v.x * norm);
  unsafeAtomicAdd(o + 1, v.y * norm);
  unsafeAtomicAdd(o + 2, v.z * norm);
  unsafeAtomicAdd(o + 3, v.w * norm);
}

// ------------------------------------------------------------ elementwise ----
__global__ void __launch_bounds__(256) gcn_relu(float* __restrict__ a, int n4) {
  int i = blockIdx.x * 256 + threadIdx.x;
  if (i < n4) {
    float4 v = ((float4*)a)[i];
    v.x = fmaxf(v.x, 0.f); v.y = fmaxf(v.y, 0.f);
    v.z = fmaxf(v.z, 0.f); v.w = fmaxf(v.w, 0.f);
    ((float4*)a)[i] = v;
  }
}

__global__ void __launch_bounds__(256) gcn_dup(const float* __restrict__ in,
                                               float* __restrict__ out, int n4) {
  int i = blockIdx.x * 256 + threadIdx.x;
  if (i < n4) ((float4*)out)[i] = ((const float4*)in)[i];
}

// ---------------------------------------------------------------- launcher ---
extern "C" void kernel_launch(void* const* d_in, const int* in_sizes, int n_in,
                              void* d_out, int out_size, void* d_ws, size_t ws_size,
                              hipStream_t stream) {
  const float*     x   = (const float*)d_in[0];
  const long long* ei  = (const long long*)d_in[1];   // int64 [2, E]
  const float*     W1  = (const float*)d_in[2];
  const float*     b1  = (const float*)d_in[3];
  const float*     W2  = (const float*)d_in[4];
  const float*     b2  = (const float*)d_in[5];

  const int N = in_sizes[0] / DMODEL;
  const int E = in_sizes[1] / 2;
  const long long* src = ei;
  const long long* dst = ei + E;

  // Workspace layout: dinv[N] | hw[N*128] | agg1[N*128]
  float* dinv = (float*)d_ws;
  size_t npad = ((size_t)N + 63) & ~(size_t)63;
  float* hw   = dinv + npad;
  float* agg1 = hw + (size_t)N * DMODEL;

  float* out0 = (float*)d_out;
  float* out1 = out0 + (size_t)N * DMODEL;

  const int ntiles   = (N + 15) / 16;
  const int ggrid    = ntiles < 4096 ? ntiles : 4096;
  const unsigned gN  = (unsigned)((N + 255) / 256);
  const unsigned gE  = (unsigned)((E + 255) / 256);
  const unsigned gSc = (unsigned)(((size_t)E * 32 + 255) / 256);
  const int n4       = N * (DMODEL / 4);
  const unsigned g4  = (unsigned)((n4 + 255) / 256);

  // degrees -> dinv
  gcn_deg_init <<<gN, 256, 0, stream>>>(dinv, N);
  gcn_deg_accum<<<gE, 256, 0, stream>>>(dst, dinv, E);
  gcn_dinv     <<<gN, 256, 0, stream>>>(dinv, N);

  // layer 1: hw1 = x@W1 ; agg1 = hw1*dinv^2 + b1 ; agg1 += scatter ; relu
  gcn_gemm_fused<<<ggrid, 256, 0, stream>>>(x, W1, b1, dinv, hw, agg1, N, ntiles);
  gcn_scatter   <<<gSc, 256, 0, stream>>>(src, dst, dinv, hw, agg1, E);
  gcn_relu      <<<g4, 256, 0, stream>>>(agg1, n4);

  // layer 2: hw2 = h@W2 ; out0 = hw2*dinv^2 + b2 ; out0 += scatter
  gcn_gemm_fused<<<ggrid, 256, 0, stream>>>(agg1, W2, b2, dinv, hw, out0, N, ntiles);
  gcn_scatter   <<<gSc, 256, 0, stream>>>(src, dst, dinv, hw, out0, E);

  // tuple (out, out)
  gcn_dup<<<g4, 256, 0, stream>>>(out0, out1, n4);
}